// PredictionLayer_55490977464949
// MI455X (gfx1250) — compile-verified
//
#include <hip/hip_runtime.h>
#include <math.h>

typedef float v2f __attribute__((ext_vector_type(2)));
typedef float v8f __attribute__((ext_vector_type(8)));

#define IN_F 128
#define WAVES_PER_BLOCK 4
#define ROW_STRIDE 132   // 128 + 4 pad floats -> conflict-free strided LDS reads
#define KSTEPS (IN_F / 4)

// ---------------------------------------------------------------------------
// Phase 1: per-node projections  S[n] = dot(X[n], W[0:128]),
//                                T[n] = dot(X[n], W[128:256])
// Formulated as (16 nodes x 128) @ (128 x 16) with only B columns 0/1 nonzero,
// accumulated by 32 chained V_WMMA_F32_16X16X4_F32 ops per 16-node tile.
// B fragments are pre-materialized per-lane in LDS so the hot loop is just
// 2x ds_load_b64 + v_wmma with no divergence.
// ---------------------------------------------------------------------------
__global__ __launch_bounds__(WAVES_PER_BLOCK * 32)
void node_proj_kernel(const float* __restrict__ X, const float* __restrict__ W,
                      float* __restrict__ S, float* __restrict__ T,
                      int ntiles, int n_nodes)
{
    // Bl[i][lane][0..1]: exact WMMA B fragment for K-step i, per lane.
    __shared__ __align__(16) float Bl[KSTEPS * 32 * 2];                // 8 KB
    __shared__ __align__(16) float Xs[WAVES_PER_BLOCK][16 * ROW_STRIDE]; // ~33 KB

    const int tid  = threadIdx.x;
    const int wid  = tid >> 5;
    const int lane = tid & 31;
    const int col  = lane & 15;   // N index of D / B
    const int half = lane >> 4;   // K-half for A/B fragments, M-half for D

    // One-time B-fragment table build (select logic lives here, not in the loop).
    // Fragment for (kstep i, lane l): b = { B[4i+2h][c], B[4i+2h+1][c] } where
    // B[k][0] = W_src[k], B[k][1] = W_trg[k], else 0.
    for (int idx = tid; idx < KSTEPS * 32; idx += WAVES_PER_BLOCK * 32) {
        const int i = idx >> 5;
        const int l = idx & 31;
        const int c = l & 15;
        const int k0 = 4 * i + 2 * (l >> 4);
        float bx = 0.0f, by = 0.0f;
        if (c == 0)      { bx = W[k0];        by = W[k0 + 1];        }
        else if (c == 1) { bx = W[IN_F + k0]; by = W[IN_F + k0 + 1]; }
        Bl[idx * 2]     = bx;
        Bl[idx * 2 + 1] = by;
    }

    int tile = blockIdx.x * WAVES_PER_BLOCK + wid;
    if (tile >= ntiles) tile = ntiles - 1;  // tail waves redo last tile (uniform barriers)
    const int node_base = tile * 16;

    // Stage this wave's 16 node rows: one coalesced b128 load per row.
    for (int r = 0; r < 16; ++r) {
        int row = node_base + r;
        if (row >= n_nodes) row = n_nodes - 1;
        const float4 v = *reinterpret_cast<const float4*>(X + (size_t)row * IN_F + lane * 4);
        *reinterpret_cast<float4*>(&Xs[wid][r * ROW_STRIDE + lane * 4]) = v;
    }
    __syncthreads();

    v8f c = {};
    const float* xrow  = &Xs[wid][col * ROW_STRIDE + 2 * half];  // A row M = col
    const float* bfrag = &Bl[lane * 2];

    #pragma unroll
    for (int i = 0; i < KSTEPS; ++i) {
        const float2 av = *reinterpret_cast<const float2*>(xrow + 4 * i);
        const float2 bv = *reinterpret_cast<const float2*>(bfrag + 64 * i);
        v2f a; a.x = av.x; a.y = av.y;
        v2f b; b.x = bv.x; b.y = bv.y;
        c = __builtin_amdgcn_wmma_f32_16x16x4_f32(false, a, false, b,
                                                  (short)0, c, false, false);
    }

    // D layout: lanes 0-15 vgpr v -> M=v ; lanes 16-31 vgpr v -> M=8+v ; N = lane%16
    const int mbase = node_base + 8 * half;
    if (col == 0) {
        #pragma unroll
        for (int v = 0; v < 8; ++v) { int m = mbase + v; if (m < n_nodes) S[m] = c[v]; }
    } else if (col == 1) {
        #pragma unroll
        for (int v = 0; v < 8; ++v) { int m = mbase + v; if (m < n_nodes) T[m] = c[v]; }
    }
}

// ---------------------------------------------------------------------------
// Phase 2: per-edge gather + sigmoid. S/T (800KB) live in L2.
// ---------------------------------------------------------------------------
__global__ void edge_kernel(const float* __restrict__ S, const float* __restrict__ T,
                            const int* __restrict__ src, const int* __restrict__ trg,
                            const float* __restrict__ bias, float* __restrict__ out,
                            int n_edges)
{
    int i = blockIdx.x * blockDim.x + threadIdx.x;
    if (i < n_edges) {
        float z = S[src[i]] + T[trg[i]] + bias[0];
        out[i] = 1.0f / (1.0f + __expf(-z));
    }
}

// Fallback if workspace is too small: direct per-edge dot products.
__global__ void edge_naive_kernel(const float* __restrict__ X, const float* __restrict__ W,
                                  const int* __restrict__ src, const int* __restrict__ trg,
                                  const float* __restrict__ bias, float* __restrict__ out,
                                  int n_edges)
{
    int i = blockIdx.x * blockDim.x + threadIdx.x;
    if (i < n_edges) {
        const float* xs = X + (size_t)src[i] * IN_F;
        const float* xt = X + (size_t)trg[i] * IN_F;
        float z = bias[0];
        #pragma unroll 8
        for (int k = 0; k < IN_F; ++k) z += xs[k] * W[k] + xt[k] * W[IN_F + k];
        out[i] = 1.0f / (1.0f + __expf(-z));
    }
}

extern "C" void kernel_launch(void* const* d_in, const int* in_sizes, int n_in,
                              void* d_out, int out_size, void* d_ws, size_t ws_size,
                              hipStream_t stream)
{
    const float* X    = (const float*)d_in[0];
    const int*   esrc = (const int*)d_in[1];   // JAX w/o x64: int64 -> int32
    const int*   etrg = (const int*)d_in[2];
    const float* W    = (const float*)d_in[3];
    const float* b    = (const float*)d_in[4];
    const int n_nodes = in_sizes[0] / IN_F;
    const int n_edges = in_sizes[1];
    float* out = (float*)d_out;

    const size_t need = 2 * (size_t)n_nodes * sizeof(float);
    if (d_ws != nullptr && ws_size >= need) {
        float* S = (float*)d_ws;
        float* T = S + n_nodes;
        const int ntiles = (n_nodes + 15) / 16;
        const int blocks = (ntiles + WAVES_PER_BLOCK - 1) / WAVES_PER_BLOCK;
        node_proj_kernel<<<blocks, WAVES_PER_BLOCK * 32, 0, stream>>>(
            X, W, S, T, ntiles, n_nodes);
        edge_kernel<<<(n_edges + 255) / 256, 256, 0, stream>>>(
            S, T, esrc, etrg, b, out, n_edges);
    } else {
        edge_naive_kernel<<<(n_edges + 255) / 256, 256, 0, stream>>>(
            X, W, esrc, etrg, b, out, n_edges);
    }
}